// ShortConv_14104672600489
// MI455X (gfx1250) — compile-verified
//
#include <hip/hip_runtime.h>

// Problem constants (match reference setup_inputs()):
#define DIMC   2048   // channels (innermost, contiguous)
#define SEQL   4096   // sequence length
#define NBATCH 4
#define KW     4      // conv kernel taps (causal, left pad KW-1)

#define LCHUNK 32     // rows of L per thread (halo reload = 3/32 ≈ 9% extra x reads)
#define TPB    256    // threads per block (8 wave32s)
#define PFDIST 12     // prefetch distance in L rows (~12*128B per wave-line stream)

typedef float v4f __attribute__((ext_vector_type(4)));

__device__ __forceinline__ float silu_f(float v) {
  // v * sigmoid(v); fast path: v_exp_f32 + v_rcp_f32.
  // v -> -inf: exp -> inf, rcp -> 0, result 0 (correct limit).
  return v * __builtin_amdgcn_rcpf(1.0f + __expf(-v));
}

__global__ __launch_bounds__(TPB, 1) void dwconv4_silu_kernel(
    const float* __restrict__ x,   // (B, L, D)
    const float* __restrict__ w,   // (D, 1, K) -> flat d*K + k
    float* __restrict__ y)         // (B, L, D)
{
  const int d4 = blockIdx.x * TPB + threadIdx.x;   // which float4 column of D
  const int l0 = blockIdx.y * LCHUNK;              // chunk start in L
  const int b  = blockIdx.z;
  const int d  = d4 << 2;

  // Per-channel taps: w[(d+i)*KW .. +3] is one contiguous float4 per channel.
  const v4f w0 = *(const v4f*)(w + (size_t)(d + 0) * KW);
  const v4f w1 = *(const v4f*)(w + (size_t)(d + 1) * KW);
  const v4f w2 = *(const v4f*)(w + (size_t)(d + 2) * KW);
  const v4f w3 = *(const v4f*)(w + (size_t)(d + 3) * KW);

  const float* xp = x + (size_t)b * SEQL * DIMC + d;
  float*       yp = y + (size_t)b * SEQL * DIMC + d;

  // Sliding window registers: x rows l-3, l-2, l-1 (float4 each).
  v4f xm3 = (v4f)0.0f, xm2 = (v4f)0.0f, xm1 = (v4f)0.0f;
  if (l0 >= 3) {  // l0 is a multiple of LCHUNK: either 0 (causal zero pad) or >= 3
    xm3 = __builtin_nontemporal_load((const v4f*)(xp + (size_t)(l0 - 3) * DIMC));
    xm2 = __builtin_nontemporal_load((const v4f*)(xp + (size_t)(l0 - 2) * DIMC));
    xm1 = __builtin_nontemporal_load((const v4f*)(xp + (size_t)(l0 - 1) * DIMC));
  }

#pragma unroll 4
  for (int l = l0; l < l0 + LCHUNK; ++l) {
    // Stream the current row (one b128 load per thread; x read exactly once).
    v4f xc = __builtin_nontemporal_load((const v4f*)(xp + (size_t)l * DIMC));

    // gfx1250 prefetch of the stream PFDIST rows ahead (clamped in-bounds).
    int lp = l + PFDIST;
    lp = lp < SEQL ? lp : (SEQL - 1);
    __builtin_prefetch(xp + (size_t)lp * DIMC, 0, 1);

    // y[l] = w[k=0]*x[l-3] + w[k=1]*x[l-2] + w[k=2]*x[l-1] + w[k=3]*x[l]
    v4f v;
    v.x = w0.x * xm3.x + w0.y * xm2.x + w0.z * xm1.x + w0.w * xc.x;
    v.y = w1.x * xm3.y + w1.y * xm2.y + w1.z * xm1.y + w1.w * xc.y;
    v.z = w2.x * xm3.z + w2.y * xm2.z + w2.z * xm1.z + w2.w * xc.z;
    v.w = w3.x * xm3.w + w3.y * xm2.w + w3.z * xm1.w + w3.w * xc.w;

    v.x = silu_f(v.x);
    v.y = silu_f(v.y);
    v.z = silu_f(v.z);
    v.w = silu_f(v.w);

    __builtin_nontemporal_store(v, (v4f*)(yp + (size_t)l * DIMC));

    // Slide the causal window.
    xm3 = xm2; xm2 = xm1; xm1 = xc;
  }
}

extern "C" void kernel_launch(void* const* d_in, const int* in_sizes, int n_in,
                              void* d_out, int out_size, void* d_ws, size_t ws_size,
                              hipStream_t stream) {
  (void)in_sizes; (void)n_in; (void)out_size; (void)d_ws; (void)ws_size;
  const float* x = (const float*)d_in[0];   // (4, 4096, 2048) fp32
  const float* w = (const float*)d_in[1];   // (2048, 1, 4)    fp32
  float*       y = (float*)d_out;           // (4, 4096, 2048) fp32

  dim3 grid((DIMC / 4) / TPB,   // 2 blocks across channels
            SEQL / LCHUNK,      // 128 L-chunks
            NBATCH);            // 4 batches  -> 1024 blocks, 8192 wave32s
  dwconv4_silu_kernel<<<grid, dim3(TPB, 1, 1), 0, stream>>>(x, w, y);
}